// AV_VQVAE_Encoder_87668872446316
// MI455X (gfx1250) — compile-verified
//
#include <hip/hip_runtime.h>

// CDNA5 / gfx1250, wave32. Full-precision V_WMMA_F32_16X16X4_F32 for both the
// distance GEMM (argmin-sensitive) and the cmcm log-prob GEMM, with software-
// pipelined register double-buffering so loads overlap WMMAs.

typedef float v2f __attribute__((ext_vector_type(2)));
typedef float v8f __attribute__((ext_vector_type(8)));

__device__ __forceinline__ v8f wmma_f32x4(v2f a, v2f b, v8f c) {
    // 8-arg pattern: (neg_a, A, neg_b, B, c_mod, C, reuse_a, reuse_b)
    return __builtin_amdgcn_wmma_f32_16x16x4_f32(false, a, false, b, (short)0, c,
                                                 false, false);
}

// ---------------------------------------------------------------------------
// 1) row-wise sum of squares (D == 256 == blockDim.x)
// ---------------------------------------------------------------------------
__global__ void rowsumsq_kernel(const float* __restrict__ X, float* __restrict__ out,
                                int D) {
    __shared__ float sm[256];
    const int row = blockIdx.x;
    const int tid = threadIdx.x;
    float v = X[(size_t)row * D + tid];
    sm[tid] = v * v;
    __syncthreads();
    for (int s = 128; s > 0; s >>= 1) {
        if (tid < s) sm[tid] += sm[tid + s];
        __syncthreads();
    }
    if (tid == 0) out[row] = sm[0];
}

// ---------------------------------------------------------------------------
// 2) dist = ||x||^2 + ||e||^2 - 2 * X @ E^T   via V_WMMA_F32_16X16X4_F32
//    one wave -> 16(n) x 64(m) strip; A fragment reused across 4 B tiles.
//    K pipelined in chunks of 16 with register double-buffering: next chunk's
//    20 b64 loads are in flight while 16 WMMAs consume the current chunk.
// ---------------------------------------------------------------------------
template<int D, int M>
__global__ void dist_kernel(const float* __restrict__ X, const float* __restrict__ E,
                            const float* __restrict__ x2, const float* __restrict__ e2,
                            float* __restrict__ dist, int N) {
    const int lane = threadIdx.x & 31;
    const int wid  = (blockIdx.x * blockDim.x + threadIdx.x) >> 5;
    const int mQuads = M >> 6;                 // 64-wide m strips
    const int nt = wid / mQuads;
    const int mq = wid - nt * mQuads;
    if (nt * 16 >= N) return;                  // wave-uniform
    const int n0 = nt << 4;
    const int m0 = mq << 6;

    const int r   = lane & 15;                 // row within tile (A) / col (B)
    const int kgr = (lane >> 4) << 1;          // 0 or 2: k-pair base

    const float* ap = X + (size_t)(n0 + r) * D + kgr;
    const float* bp0 = E + (size_t)(m0 +  0 + r) * D + kgr;
    const float* bp1 = E + (size_t)(m0 + 16 + r) * D + kgr;
    const float* bp2 = E + (size_t)(m0 + 32 + r) * D + kgr;
    const float* bp3 = E + (size_t)(m0 + 48 + r) * D + kgr;

    const v8f vz = {0.f,0.f,0.f,0.f,0.f,0.f,0.f,0.f};
    v8f acc0 = vz, acc1 = vz, acc2 = vz, acc3 = vz;

    v2f aC[4], b0C[4], b1C[4], b2C[4], b3C[4];
    #pragma unroll
    for (int s = 0; s < 4; ++s) {              // prologue: chunk 0 (k = 0..15)
        aC[s]  = *(const v2f*)(ap  + 4 * s);
        b0C[s] = *(const v2f*)(bp0 + 4 * s);
        b1C[s] = *(const v2f*)(bp1 + 4 * s);
        b2C[s] = *(const v2f*)(bp2 + 4 * s);
        b3C[s] = *(const v2f*)(bp3 + 4 * s);
    }

    #pragma unroll
    for (int kk = 0; kk < D; kk += 16) {
        v2f aN[4], b0N[4], b1N[4], b2N[4], b3N[4];
        if (kk + 16 < D) {                     // issue next chunk's loads early
            #pragma unroll
            for (int s = 0; s < 4; ++s) {
                const int k = kk + 16 + 4 * s;
                aN[s]  = *(const v2f*)(ap  + k);
                b0N[s] = *(const v2f*)(bp0 + k);
                b1N[s] = *(const v2f*)(bp1 + k);
                b2N[s] = *(const v2f*)(bp2 + k);
                b3N[s] = *(const v2f*)(bp3 + k);
            }
        }
        #pragma unroll
        for (int s = 0; s < 4; ++s) {
            acc0 = wmma_f32x4(aC[s], b0C[s], acc0);
            acc1 = wmma_f32x4(aC[s], b1C[s], acc1);
            acc2 = wmma_f32x4(aC[s], b2C[s], acc2);
            acc3 = wmma_f32x4(aC[s], b3C[s], acc3);
        }
        if (kk + 16 < D) {
            #pragma unroll
            for (int s = 0; s < 4; ++s) {
                aC[s] = aN[s]; b0C[s] = b0N[s]; b1C[s] = b1N[s];
                b2C[s] = b2N[s]; b3C[s] = b3N[s];
            }
        }
    }

    const int nrow = n0 + ((lane >> 4) << 3);  // D layout: row = v + (lane/16)*8
    const float em0 = e2[m0 +  0 + r];
    const float em1 = e2[m0 + 16 + r];
    const float em2 = e2[m0 + 32 + r];
    const float em3 = e2[m0 + 48 + r];
    #pragma unroll
    for (int v = 0; v < 8; ++v) {
        const int n  = nrow + v;
        const float xn = x2[n];
        size_t base = (size_t)n * M + m0 + r;
        dist[base +  0] = xn + em0 - 2.0f * acc0[v];
        dist[base + 16] = xn + em1 - 2.0f * acc1[v];
        dist[base + 32] = xn + em2 - 2.0f * acc2[v];
        dist[base + 48] = xn + em3 - 2.0f * acc3[v];
    }
}

// ---------------------------------------------------------------------------
// 3) argmin over M per row (first-occurrence ties) + gather embedding row
// ---------------------------------------------------------------------------
__global__ void argmin_gather_kernel(const float* __restrict__ dist,
                                     const float* __restrict__ E,
                                     float* __restrict__ out, int M, int D) {
    __shared__ float sv[256];
    __shared__ int   si[256];
    const int row = blockIdx.x;
    const int tid = threadIdx.x;
    float best = 3.402823466e+38f;
    int   bidx = 0x7fffffff;
    for (int i = tid; i < M; i += 256) {
        float d = dist[(size_t)row * M + i];
        if (d < best) { best = d; bidx = i; }
    }
    sv[tid] = best; si[tid] = bidx;
    __syncthreads();
    for (int s = 128; s > 0; s >>= 1) {
        if (tid < s) {
            float ov = sv[tid + s]; int oi = si[tid + s];
            if (ov < sv[tid] || (ov == sv[tid] && oi < si[tid])) {
                sv[tid] = ov; si[tid] = oi;
            }
        }
        __syncthreads();
    }
    const int e = si[0];
    out[(size_t)row * D + tid] = E[(size_t)e * D + tid];   // D == 256 == blockDim
}

// ---------------------------------------------------------------------------
// 4) dual-temperature softmax: s = sqrt(max(d,0));
//    p1 = softmax(-s) (t=1.0) -> logp out;  adj = softmax(-2s) (t=0.5) -> in-place
//    exp(2z) == exp(z)^2 so a single exp serves both.
// ---------------------------------------------------------------------------
__global__ void softmax_rows_kernel(float* __restrict__ dist,
                                    float* __restrict__ logp, int M) {
    __shared__ float sm[256];
    __shared__ float sred[3];
    const int row = blockIdx.x;
    const int tid = threadIdx.x;
    const size_t base = (size_t)row * M;

    float s[4];
    #pragma unroll
    for (int i = 0; i < 4; ++i) {
        float d = dist[base + tid + i * 256];
        s[i] = sqrtf(fmaxf(d, 0.0f));
    }
    float lmin = fminf(fminf(s[0], s[1]), fminf(s[2], s[3]));
    sm[tid] = lmin;
    __syncthreads();
    for (int st = 128; st > 0; st >>= 1) {
        if (tid < st) sm[tid] = fminf(sm[tid], sm[tid + st]);
        __syncthreads();
    }
    if (tid == 0) sred[0] = sm[0];
    __syncthreads();
    const float smin = sred[0];
    __syncthreads();

    float e1[4];
    float l1 = 0.f, l2 = 0.f;
    #pragma unroll
    for (int i = 0; i < 4; ++i) {
        float z = smin - s[i];          // <= 0
        e1[i] = expf(z);
        l1 += e1[i];
        l2 += e1[i] * e1[i];
    }
    sm[tid] = l1;
    __syncthreads();
    for (int st = 128; st > 0; st >>= 1) {
        if (tid < st) sm[tid] += sm[tid + st];
        __syncthreads();
    }
    if (tid == 0) sred[1] = sm[0];
    __syncthreads();
    sm[tid] = l2;
    __syncthreads();
    for (int st = 128; st > 0; st >>= 1) {
        if (tid < st) sm[tid] += sm[tid + st];
        __syncthreads();
    }
    if (tid == 0) sred[2] = sm[0];
    __syncthreads();

    const float inv1 = 1.0f / sred[1];
    const float inv2 = 1.0f / sred[2];
    #pragma unroll
    for (int i = 0; i < 4; ++i) {
        float p1  = e1[i] * inv1;
        float adj = (e1[i] * e1[i]) * inv2;
        dist[base + tid + i * 256] = adj;                 // adj (t=0.5) in place
        logp[base + tid + i * 256] = logf(p1 + 1e-10f);   // log(pH + 1e-10)
    }
}

// ---------------------------------------------------------------------------
// 5) S[t] = adj[t] (64xM) @ logq[t]^T (Mx64); rows of slice t live at flat
//    row n = b*T + t in the [B*T, M] matrices. f32 WMMA, pipelined K chunks.
// ---------------------------------------------------------------------------
template<int M>
__global__ void cmcm_gemm_kernel(const float* __restrict__ Aadj,
                                 const float* __restrict__ Blogq,
                                 float* __restrict__ S, int T, int Bb) {
    const int lane = threadIdx.x & 31;
    const int wid  = (blockIdx.x * blockDim.x + threadIdx.x) >> 5;
    const int bt = Bb >> 4;                    // 4 tiles per dim
    const int tilesPerT = bt * bt;             // 16
    const int t = wid / tilesPerT;
    if (t >= T) return;                        // wave-uniform
    const int tile = wid - t * tilesPerT;
    const int b0 = (tile / bt) << 4;
    const int c0 = (tile % bt) << 4;

    const int r   = lane & 15;
    const int kgr = (lane >> 4) << 1;

    const float* ap = Aadj  + ((size_t)(b0 + r) * T + t) * M + kgr;
    const float* bp = Blogq + ((size_t)(c0 + r) * T + t) * M + kgr;

    v8f acc = {0.f,0.f,0.f,0.f,0.f,0.f,0.f,0.f};

    v2f aC[4], bC[4];
    #pragma unroll
    for (int s = 0; s < 4; ++s) {              // prologue: chunk 0
        aC[s] = *(const v2f*)(ap + 4 * s);
        bC[s] = *(const v2f*)(bp + 4 * s);
    }
    #pragma unroll 4
    for (int kk = 0; kk < M; kk += 16) {
        v2f aN[4], bN[4];
        if (kk + 16 < M) {
            #pragma unroll
            for (int s = 0; s < 4; ++s) {
                aN[s] = *(const v2f*)(ap + kk + 16 + 4 * s);
                bN[s] = *(const v2f*)(bp + kk + 16 + 4 * s);
            }
        }
        #pragma unroll
        for (int s = 0; s < 4; ++s)
            acc = wmma_f32x4(aC[s], bC[s], acc);
        if (kk + 16 < M) {
            #pragma unroll
            for (int s = 0; s < 4; ++s) { aC[s] = aN[s]; bC[s] = bN[s]; }
        }
    }

    const int brow = b0 + ((lane >> 4) << 3);
    #pragma unroll
    for (int v = 0; v < 8; ++v)
        S[((size_t)t * Bb + brow + v) * Bb + c0 + r] = acc[v];
}

// ---------------------------------------------------------------------------
// 6) global max(-S): grid-stride partials then single-block finalize
// ---------------------------------------------------------------------------
__global__ void negmax_partial_kernel(const float* __restrict__ S,
                                      float* __restrict__ partial, int n) {
    __shared__ float sm[256];
    const int tid = threadIdx.x;
    float m = -3.402823466e+38f;
    for (int i = blockIdx.x * 256 + tid; i < n; i += gridDim.x * 256)
        m = fmaxf(m, -S[i]);
    sm[tid] = m;
    __syncthreads();
    for (int s = 128; s > 0; s >>= 1) {
        if (tid < s) sm[tid] = fmaxf(sm[tid], sm[tid + s]);
        __syncthreads();
    }
    if (tid == 0) partial[blockIdx.x] = sm[0];
}

__global__ void max_final_kernel(const float* __restrict__ p1,
                                 const float* __restrict__ p2,
                                 float* __restrict__ mx) {
    __shared__ float sm[256];
    const int tid = threadIdx.x;
    sm[tid] = p1[tid];
    __syncthreads();
    for (int s = 128; s > 0; s >>= 1) {
        if (tid < s) sm[tid] = fmaxf(sm[tid], sm[tid + s]);
        __syncthreads();
    }
    if (tid == 0) mx[0] = sm[0];
    __syncthreads();
    sm[tid] = p2[tid];
    __syncthreads();
    for (int s = 128; s > 0; s >>= 1) {
        if (tid < s) sm[tid] = fmaxf(sm[tid], sm[tid + s]);
        __syncthreads();
    }
    if (tid == 0) mx[1] = sm[0];
}

// ---------------------------------------------------------------------------
// 7) per-t: sum_b log( exp(S[t,b,b]+mx) / (sum_c exp(S[t,b,c]+mx) + EPS) )
// ---------------------------------------------------------------------------
__global__ void loss_rows_kernel(const float* __restrict__ S,
                                 const float* __restrict__ mx, int which,
                                 float* __restrict__ tPartial, int Bb) {
    __shared__ float sm[64];
    const int t = blockIdx.x;
    const int b = threadIdx.x;                 // 64 threads
    const float m = mx[which];
    const size_t base = ((size_t)t * Bb + b) * Bb;
    float sum = 0.f;
    for (int c = 0; c < Bb; ++c) sum += expf(S[base + c] + m);
    float diag = expf(S[base + b] + m);
    float term = logf(diag / (sum + 1e-5f));
    sm[b] = term;
    __syncthreads();
    for (int s = 32; s > 0; s >>= 1) {
        if (b < s) sm[b] += sm[b + s];
        __syncthreads();
    }
    if (b == 0) tPartial[which * gridDim.x + t] = sm[0];
}

__global__ void loss_final_kernel(const float* __restrict__ tPartial,
                                  float* __restrict__ out, int T, int Bb) {
    __shared__ float sm[128];
    __shared__ float P[2];
    const int tid = threadIdx.x;               // 128 == T
    sm[tid] = tPartial[tid];
    __syncthreads();
    for (int s = 64; s > 0; s >>= 1) {
        if (tid < s) sm[tid] += sm[tid + s];
        __syncthreads();
    }
    if (tid == 0) P[0] = sm[0];
    __syncthreads();
    sm[tid] = tPartial[T + tid];
    __syncthreads();
    for (int s = 64; s > 0; s >>= 1) {
        if (tid < s) sm[tid] += sm[tid + s];
        __syncthreads();
    }
    if (tid == 0) {
        P[1] = sm[0];
        float inv = 1.0f / (float)(T * Bb);
        float L1 = -P[0] * inv;
        float L2 = -P[1] * inv;
        out[0] = 0.5f * (L1 + L2);
    }
}

// ---------------------------------------------------------------------------
extern "C" void kernel_launch(void* const* d_in, const int* in_sizes, int n_in,
                              void* d_out, int out_size, void* d_ws, size_t ws_size,
                              hipStream_t stream) {
    const float* audio = (const float*)d_in[0];   // [B,T,D]
    const float* video = (const float*)d_in[1];   // [B,T,D]
    const float* emb   = (const float*)d_in[2];   // [M,D]
    float* out = (float*)d_out;                   // [loss][a_quant][v_quant]

    const int Bb = 64, T = 128;
    constexpr int D = 256, M = 1024;
    const int N = Bb * T;                         // 8192 rows

    float* ws = (float*)d_ws;
    size_t off = 0;
    float* e2     = ws + off; off += M;
    float* a2     = ws + off; off += N;
    float* v2     = ws + off; off += N;
    float* dist_a = ws + off; off += (size_t)N * M;   // -> adj_a in place
    float* dist_v = ws + off; off += (size_t)N * M;   // -> adj_v in place
    float* logp_a = ws + off; off += (size_t)N * M;
    float* logp_v = ws + off; off += (size_t)N * M;
    float* S1     = ws + off; off += (size_t)T * Bb * Bb;
    float* S2     = ws + off; off += (size_t)T * Bb * Bb;
    float* part1  = ws + off; off += 256;
    float* part2  = ws + off; off += 256;
    float* mx     = ws + off; off += 2;
    float* tPart  = ws + off; off += 2 * T;

    // 1) norms
    rowsumsq_kernel<<<M, 256, 0, stream>>>(emb,   e2, D);
    rowsumsq_kernel<<<N, 256, 0, stream>>>(audio, a2, D);
    rowsumsq_kernel<<<N, 256, 0, stream>>>(video, v2, D);

    // 2) distance GEMMs (WMMA f32): (N/16)*(M/64) waves, 4 waves/block
    const int distWaves  = (N / 16) * (M / 64);   // 8192
    const int distBlocks = distWaves / 4;         // 2048
    dist_kernel<D, M><<<distBlocks, 128, 0, stream>>>(audio, emb, a2, e2, dist_a, N);
    dist_kernel<D, M><<<distBlocks, 128, 0, stream>>>(video, emb, v2, e2, dist_v, N);

    // 3) argmin + straight-through quantized outputs (== gathered codebook rows)
    argmin_gather_kernel<<<N, 256, 0, stream>>>(dist_a, emb, out + 1, M, D);
    argmin_gather_kernel<<<N, 256, 0, stream>>>(dist_v, emb, out + 1 + (size_t)N * D, M, D);

    // 4) dual-temperature softmax (adj in place of dist, log(pH) separate)
    softmax_rows_kernel<<<N, 256, 0, stream>>>(dist_a, logp_a, M);
    softmax_rows_kernel<<<N, 256, 0, stream>>>(dist_v, logp_v, M);

    // 5) cmcm GEMMs (WMMA f32): T * 16 tiles per pairing, 4 waves/block
    const int cmWaves  = T * (Bb / 16) * (Bb / 16);  // 2048
    const int cmBlocks = cmWaves / 4;                // 512
    cmcm_gemm_kernel<M><<<cmBlocks, 128, 0, stream>>>(dist_a, logp_v, S1, T, Bb);
    cmcm_gemm_kernel<M><<<cmBlocks, 128, 0, stream>>>(dist_v, logp_a, S2, T, Bb);

    // 6) global max(-S)
    const int Sn = T * Bb * Bb;
    negmax_partial_kernel<<<256, 256, 0, stream>>>(S1, part1, Sn);
    negmax_partial_kernel<<<256, 256, 0, stream>>>(S2, part2, Sn);
    max_final_kernel<<<1, 256, 0, stream>>>(part1, part2, mx);

    // 7) per-t loss rows, then final scalar
    loss_rows_kernel<<<T, 64, 0, stream>>>(S1, mx, 0, tPart, Bb);
    loss_rows_kernel<<<T, 64, 0, stream>>>(S2, mx, 1, tPart, Bb);
    loss_final_kernel<<<1, 128, 0, stream>>>(tPart, out, T, Bb);
}